// GConv_47622597378608
// MI455X (gfx1250) — compile-verified
//
#include <hip/hip_runtime.h>

// ---- CDNA5 (gfx1250) wave32 WMMA types ----
typedef __attribute__((ext_vector_type(2))) float v2f;
typedef __attribute__((ext_vector_type(8))) float v8f;

#define D 128         // D_IN == UNITS == 128
#define NPITCH 144    // float2 pitch per K-pair row of w: 2*144 % 64 == 32 -> half-waves
                      // land on disjoint 32-bank halves, conflict-free b64 loads
#define XPAD 132      // float pitch per x-tile row: (4L + k0) banks, conflict-free b64 loads

// -------------------------------------------------------------------------
// Kernel 1: zero the output accumulator (float4 stores)
// -------------------------------------------------------------------------
__global__ __launch_bounds__(256) void GConv_zero(float4* __restrict__ out, int n4) {
    int i = blockIdx.x * 256 + threadIdx.x;
    if (i < n4) out[i] = make_float4(0.f, 0.f, 0.f, 0.f);
}

// -------------------------------------------------------------------------
// Kernel 2: h = x @ w via V_WMMA_F32_16X16X4_F32 (full fp32 precision).
// One wave per 16-row block of x; 8 N-tiles of 16x16 accumulated per wave.
// w staged in LDS K-pair interleaved: ldsW[kp][n] = {w[2kp][n], w[2kp+1][n]}
// so each B fragment is ONE aligned ds_load_b64 into a contiguous VGPR pair
// (kills the v_dual_mov repacking seen in the previous build).
// -------------------------------------------------------------------------
__global__ __launch_bounds__(256) void GConv_gemm(const float* __restrict__ x,
                                                  const float* __restrict__ w,
                                                  float* __restrict__ h,
                                                  int n_nodes) {
    __shared__ v2f   ldsW[(D / 2) * NPITCH];    // 64*144*8 = 73,728 B
    __shared__ float ldsX[8][16 * XPAD];        // 8 waves * 8,448 B = 67,584 B

    const int tid  = threadIdx.x;
    const int wave = tid >> 5;
    const int lane = tid & 31;
    const int l15  = lane & 15;
    const int half = lane >> 4;                 // 0: lanes 0-15, 1: lanes 16-31

    // Cooperative load of w (128x128) into K-pair-interleaved LDS.
    // Element (r, c) -> ldsW[(r>>1)*NPITCH + c], component (r&1).
    for (int i = tid; i < D * D; i += 256) {
        const int r = i >> 7, c = i & 127;
        ((float*)&ldsW[(r >> 1) * NPITCH + c])[r & 1] = w[i];
    }

    const int rowblock = blockIdx.x * 8 + wave;        // wave-uniform
    const bool active  = (rowblock * 16) < n_nodes;

    if (active) {
        // Stage this wave's 16x128 x-tile (coalesced float4) into padded LDS.
        const float4* xt = (const float4*)(x + (size_t)rowblock * 16 * D);
        float* lx = &ldsX[wave][0];
        #pragma unroll
        for (int i = lane; i < 512; i += 32) {         // 512 float4 = 16 rows * 32
            const int r  = i >> 5;
            const int c4 = i & 31;
            float4 v = xt[i];
            float* dst = lx + r * XPAD + c4 * 4;
            dst[0] = v.x; dst[1] = v.y; dst[2] = v.z; dst[3] = v.w;
        }
    }
    __syncthreads();
    if (!active) return;                               // whole-wave exit: EXEC stays all-1s

    v8f acc[8];
    #pragma unroll
    for (int n = 0; n < 8; ++n) acc[n] = (v8f){0.f,0.f,0.f,0.f,0.f,0.f,0.f,0.f};

    const float* lx = &ldsX[wave][0];

    // K = 128 -> 32 steps of K=4. ISA A-layout: lanes 0-15 hold K{0,1}, lanes
    // 16-31 hold K{2,3}; B-layout mirrors with half-wave selecting K+2.
    // k0 = ks*4 + 2*half; K-pair index kp = k0/2 = 2*ks + half.
    #pragma unroll 4
    for (int ks = 0; ks < 32; ++ks) {
        const int k0 = ks * 4 + half * 2;
        const int kp = ks * 2 + half;
        const v2f a = *(const v2f*)(lx + l15 * XPAD + k0);   // one ds_load_b64
        const v2f* brow = &ldsW[kp * NPITCH + l15];
        #pragma unroll
        for (int n = 0; n < 8; ++n) {
            const v2f bf = brow[n * 16];                      // one ds_load_b64
            acc[n] = __builtin_amdgcn_wmma_f32_16x16x4_f32(
                         false, a, false, bf, (short)0, acc[n], false, false);
        }
    }

    // C/D layout: VGPR v -> M = v + 8*half, N = n*16 + l15.
    const int mbase = rowblock * 16 + half * 8;
    #pragma unroll
    for (int n = 0; n < 8; ++n) {
        #pragma unroll
        for (int v = 0; v < 8; ++v) {
            h[(size_t)(mbase + v) * D + n * 16 + l15] = acc[n][v];
        }
    }
}

// -------------------------------------------------------------------------
// Kernel 3: edge scatter  out[dst] += val * h[src]
// One wave32 per edge; each lane owns 4 channels (float4 gather + 4 atomics).
// h and out rows (512 B) stay L2-resident (51 MB each vs 192 MB L2).
// -------------------------------------------------------------------------
__global__ __launch_bounds__(256) void GConv_scatter(const int* __restrict__ esrc,
                                                     const int* __restrict__ edst,
                                                     const float* __restrict__ evals,
                                                     const float* __restrict__ h,
                                                     float* __restrict__ out,
                                                     int n_edges) {
    const int t = blockIdx.x * 256 + threadIdx.x;
    const int e = t >> 5;
    if (e >= n_edges) return;
    const int lane = threadIdx.x & 31;

    const int   s = esrc[e];
    const int   d = edst[e];
    const float v = evals[e];

    const float4 hv = ((const float4*)(h + (size_t)s * D))[lane];
    float* o = out + (size_t)d * D + lane * 4;
    atomicAdd(o + 0, v * hv.x);
    atomicAdd(o + 1, v * hv.y);
    atomicAdd(o + 2, v * hv.z);
    atomicAdd(o + 3, v * hv.w);
}

// -------------------------------------------------------------------------
// Kernel 4: out = relu(out + b), float4 vectorized
// -------------------------------------------------------------------------
__global__ __launch_bounds__(256) void GConv_epilogue(float4* __restrict__ out,
                                                      const float4* __restrict__ b4,
                                                      int n4) {
    int i = blockIdx.x * 256 + threadIdx.x;
    if (i >= n4) return;
    float4 v  = out[i];
    float4 bb = b4[i & 31];            // 32 float4 per 128-wide row
    v.x = fmaxf(v.x + bb.x, 0.f);
    v.y = fmaxf(v.y + bb.y, 0.f);
    v.z = fmaxf(v.z + bb.z, 0.f);
    v.w = fmaxf(v.w + bb.w, 0.f);
    out[i] = v;
}

// -------------------------------------------------------------------------
extern "C" void kernel_launch(void* const* d_in, const int* in_sizes, int n_in,
                              void* d_out, int out_size, void* d_ws, size_t ws_size,
                              hipStream_t stream) {
    const float* x     = (const float*)d_in[0];
    const int*   esrc  = (const int*)  d_in[1];
    const int*   edst  = (const int*)  d_in[2];
    const float* evals = (const float*)d_in[3];
    const float* w     = (const float*)d_in[4];
    const float* b     = (const float*)d_in[5];
    float* out = (float*)d_out;
    float* h   = (float*)d_ws;                     // 100000*128*4 = 51.2 MB scratch

    const int n_nodes = in_sizes[0] / D;
    const int n_edges = in_sizes[1];

    const int n4 = n_nodes * D / 4;

    // 1) zero the accumulator output
    GConv_zero<<<(n4 + 255) / 256, 256, 0, stream>>>((float4*)out, n4);

    // 2) dense GEMM with fp32 WMMA
    const int rowblocks = (n_nodes + 15) / 16;
    GConv_gemm<<<(rowblocks + 7) / 8, 256, 0, stream>>>(x, w, h, n_nodes);

    // 3) edge scatter-add (one wave per edge)
    const long long threads = (long long)n_edges * 32;
    GConv_scatter<<<(int)((threads + 255) / 256), 256, 0, stream>>>(
        esrc, edst, evals, h, out, n_edges);

    // 4) bias + relu epilogue
    GConv_epilogue<<<(n4 + 255) / 256, 256, 0, stream>>>(
        (float4*)out, (const float4*)b, n4);
}